// KAN_65919158059311
// MI455X (gfx1250) — compile-verified
//
#include <hip/hip_runtime.h>

typedef float v2f __attribute__((ext_vector_type(2)));
typedef float v8f __attribute__((ext_vector_type(8)));

#define WMMA4(A, B, C) \
  __builtin_amdgcn_wmma_f32_16x16x4_f32(false, (A), false, (B), (short)0, (C), false, false)

namespace {
constexpr int kD0 = 64;
constexpr int kD1 = 128;
constexpr int kD2 = 64;
constexpr int kRows = 64;            // batch rows per block
constexpr int kKB0 = kD0;            // 64 k-blocks (4 wide) for layer 0  (K0 = 256)
constexpr int kKB1 = kD1;            // 128 k-blocks for layer 1          (K1 = 512)
constexpr int kS0 = kD0 * 4 + 4;     // 260 floats: A0 LDS row stride (260 % 64 == 4 -> conflict free b64)
constexpr int kS1 = kD1 * 4 + 4;     // 516 floats: A1 LDS row stride
constexpr int kB0Floats = 8 * kKB0 * 32 * 2;   // 32768 floats = 128 KB swizzled W0
constexpr int kB1Floats = 4 * kKB1 * 32 * 2;   // 32768 floats = 128 KB swizzled W1
constexpr int kLdsFloats = kRows * kS1;        // 33024 floats = 129 KB (A1 phase is the max)
}

// ---------------------------------------------------------------------------
// Gaussian B-spline basis table, identical math to the reference (fp32).
// knots = linspace(-1.3, 1.3, 9) -> step 0.325; width = 0.65 for all i;
// center_i = knots[i+1] + 0.1625. Rows normalized by (sum + 1e-6).
// ---------------------------------------------------------------------------
__device__ __forceinline__ void bv_rows(float bv[3][5]) {
#pragma unroll
  for (int g = 0; g < 3; ++g) {
    float gr = -1.0f + (float)g;
    float s = 1e-6f;
#pragma unroll
    for (int i = 0; i < 5; ++i) {
      float center = -1.3f + 0.325f * (float)(i + 1) + 0.1625f;
      float d = (gr - center) * (1.0f / 0.65f);
      float e = expf(-d * d);
      bv[g][i] = e;
      s += e;
    }
    float inv = 1.0f / s;
#pragma unroll
    for (int i = 0; i < 5; ++i) bv[g][i] *= inv;
  }
}

// Expand one scalar into the 4 activation slots: {silu, u0, u1, u2}
// where u_g are the hat-interpolation weights over the 3-point grid.
__device__ __forceinline__ void expand4(float v, float o[4]) {
  float sg = v / (1.0f + expf(-v));            // x * sigmoid(x)
  float xc = fminf(fmaxf(v, -1.0f), 1.0f);
  float gi = xc + 1.0f;                        // in [0, 2]
  o[0] = sg;
  o[1] = fmaxf(1.0f - gi, 0.0f);
  o[2] = 1.0f - fabsf(gi - 1.0f);
  o[3] = fmaxf(gi - 1.0f, 0.0f);
}

// ---------------------------------------------------------------------------
// Prep: build fused+swizzled weights. Effective weight W[c][j], c = i*4 + s:
//   s==0 : imp*bw            (base/silu path)
//   s>=1 : imp*sw*Σ_k bv[s-1][k]*cp[i][j][k]   (bv folded into weights)
// Stored per WMMA B-fragment lane order: float2 index ((nt*KB + kb)*32 + lane)
//   lane<16  -> K = kb*4 + {0,1},  N = nt*16 + lane
//   lane>=16 -> K = kb*4 + {2,3},  N = nt*16 + lane-16
// ---------------------------------------------------------------------------
__global__ void kan_prep_w(const float* __restrict__ cp, const float* __restrict__ bw,
                           const float* __restrict__ sw, const float* __restrict__ imp,
                           float* __restrict__ Bout, int J, int I) {
  int t = blockIdx.x * blockDim.x + threadIdx.x;
  int total = (J / 16) * I * 32;
  if (t >= total) return;
  int lane = t & 31;
  int kb = (t >> 5) % I;        // k-block == input feature i
  int nt = t / (32 * I);        // output col tile
  int n = nt * 16 + (lane & 15);
  int sbase = (lane >> 4) << 1; // 0 or 2

  float bv[3][5];
  bv_rows(bv);

  int idx = kb * J + n;         // (i, j) flat
  float m = imp[idx];
  const float* c5 = cp + (size_t)idx * 5;

  float vals[2];
#pragma unroll
  for (int q = 0; q < 2; ++q) {
    int s = sbase + q;
    float v;
    if (s == 0) {
      v = m * bw[idx];
    } else {
      float acc = 0.0f;
#pragma unroll
      for (int k = 0; k < 5; ++k) acc += bv[s - 1][k] * c5[k];
      v = m * sw[idx] * acc;
    }
    vals[q] = v;
  }
  ((float2*)Bout)[t] = make_float2(vals[0], vals[1]);
}

// ---------------------------------------------------------------------------
// Fused two-layer KAN. 256 threads = 8 waves, 64 batch rows per block.
// ---------------------------------------------------------------------------
__device__ __forceinline__ void expand_store(float* smem, v8f c, int rowbase, int i) {
#pragma unroll
  for (int r = 0; r < 8; ++r) {
    float o[4];
    expand4(c[r], o);
    float* p = smem + (rowbase + r) * kS1 + i * 4;
    *(float2*)p = make_float2(o[0], o[1]);
    *(float2*)(p + 2) = make_float2(o[2], o[3]);
  }
}

__global__ void __launch_bounds__(256) kan_fused(const float* __restrict__ x,
                                                 const float* __restrict__ ws,
                                                 float* __restrict__ out) {
  __shared__ float smem[kLdsFloats];

  const int tid = threadIdx.x;
  const int lane = tid & 31;
  const int w = tid >> 5;            // wave id 0..7
  const int lr = lane & 15;
  const int koff = (lane >> 4) << 1; // 0 (lanes 0-15) or 2 (lanes 16-31)
  const int half8 = (lane >> 4) << 3;
  const long long r0 = (long long)blockIdx.x * kRows;

  const float* B0 = ws;
  const float* B1 = ws + kB0Floats;

  // ---- Phase 0: expand x tile -> A0 [64 x 260] in LDS ----
  for (int e = tid; e < kRows * kD0; e += 256) {
    int row = e >> 6;
    int i = e & 63;
    float v = x[(r0 + row) * kD0 + i];
    float o[4];
    expand4(v, o);
    float* p = smem + row * kS0 + i * 4;
    *(float2*)p = make_float2(o[0], o[1]);
    *(float2*)(p + 2) = make_float2(o[2], o[3]);
  }
  __syncthreads();

  // ---- Phase 1: GEMM1. Wave w -> col tile w; all 4 row tiles (B reused x4) ----
  v8f z = {0.f, 0.f, 0.f, 0.f, 0.f, 0.f, 0.f, 0.f};
  v8f c0 = z, c1 = z, c2 = z, c3 = z;
  {
    const float* bp = B0 + ((size_t)w * kKB0 * 32 + lane) * 2;
    const float* ap = smem + lr * kS0 + koff;
#pragma unroll 4
    for (int kb = 0; kb < kKB0; ++kb) {
      v2f b = *(const v2f*)(bp + kb * 64);
      const float* a = ap + kb * 4;
      v2f a0 = *(const v2f*)(a);
      v2f a1 = *(const v2f*)(a + 16 * kS0);
      v2f a2 = *(const v2f*)(a + 32 * kS0);
      v2f a3 = *(const v2f*)(a + 48 * kS0);
      c0 = WMMA4(a0, b, c0);
      c1 = WMMA4(a1, b, c1);
      c2 = WMMA4(a2, b, c2);
      c3 = WMMA4(a3, b, c3);
    }
  }
  __syncthreads();  // everyone done reading A0 before A1 overwrites LDS

  // ---- Phase 1.5: expand h (still in registers) -> A1 [64 x 516] in LDS ----
  {
    const int i = w * 16 + lr;  // h column owned by this lane
    expand_store(smem, c0, 0 * 16 + half8, i);
    expand_store(smem, c1, 1 * 16 + half8, i);
    expand_store(smem, c2, 2 * 16 + half8, i);
    expand_store(smem, c3, 3 * 16 + half8, i);
  }
  __syncthreads();

  // ---- Phase 2: GEMM2. Wave w -> col tile (w&3), row tiles {2*(w>>2), +1} ----
  const int nt1 = w & 3;
  const int rtb = (w >> 2) << 1;
  v8f d0 = z, d1 = z;
  {
    const float* bp = B1 + ((size_t)nt1 * kKB1 * 32 + lane) * 2;
    const float* ap = smem + (rtb * 16 + lr) * kS1 + koff;
#pragma unroll 4
    for (int kb = 0; kb < kKB1; ++kb) {
      v2f b = *(const v2f*)(bp + kb * 64);
      v2f aa = *(const v2f*)(ap + kb * 4);
      v2f ab = *(const v2f*)(ap + kb * 4 + 16 * kS1);
      d0 = WMMA4(aa, b, d0);
      d1 = WMMA4(ab, b, d1);
    }
  }

  // ---- Phase 3: store C tiles (VGPR r -> row r + half8, col = lane%16) ----
  {
    const int col = nt1 * 16 + lr;
    const long long rowbase = r0 + rtb * 16 + half8;
#pragma unroll
    for (int r = 0; r < 8; ++r) {
      out[(rowbase + r) * kD2 + col] = d0[r];
      out[(rowbase + 16 + r) * kD2 + col] = d1[r];
    }
  }
}

// ---------------------------------------------------------------------------
extern "C" void kernel_launch(void* const* d_in, const int* in_sizes, int n_in,
                              void* d_out, int out_size, void* d_ws, size_t ws_size,
                              hipStream_t stream) {
  const float* x    = (const float*)d_in[0];
  const float* cp0  = (const float*)d_in[1];
  const float* bw0  = (const float*)d_in[2];
  const float* sw0  = (const float*)d_in[3];
  const float* imp0 = (const float*)d_in[4];
  const float* cp1  = (const float*)d_in[5];
  const float* bw1  = (const float*)d_in[6];
  const float* sw1  = (const float*)d_in[7];
  const float* imp1 = (const float*)d_in[8];
  float* ws = (float*)d_ws;
  float* out = (float*)d_out;

  // Build fused+swizzled weights in workspace (256 KB total).
  // W0: (128/16)*64*32 = 16384 float2 threads; W1: (64/16)*128*32 = 16384.
  kan_prep_w<<<64, 256, 0, stream>>>(cp0, bw0, sw0, imp0, ws, kD1, kD0);
  kan_prep_w<<<64, 256, 0, stream>>>(cp1, bw1, sw1, imp1, ws + kB0Floats, kD2, kD1);

  int batch = in_sizes[0] / kD0;        // 131072
  int nblocks = batch / kRows;          // 2048
  kan_fused<<<nblocks, 256, 0, stream>>>(x, ws, out);
}